// SplineLayer_83588653515336
// MI455X (gfx1250) — compile-verified
//
#include <hip/hip_runtime.h>
#include <cstdint>

// ---- Problem constants (from reference) ----
#define NFEAT      512
#define NKNOT      64
#define XMIN_F     (-3.0f)
#define XMAX_F     (3.0f)
// t = (x - XMIN) / (XMAX - XMIN) * (K-1)  ->  (x + 3) * 10.5
#define T_SCALE    (10.5f)

// ---- Tiling ----
#define BLOCK_THREADS   256          // 8 waves (wave32)
#define WAVES_PER_BLOCK 8
#define ROWS_PER_WAVE   16
#define ROWS_PER_BLOCK  (WAVES_PER_BLOCK * ROWS_PER_WAVE)   // 128

// LDS layout: coeffs[f][k] stored at float index f*LDS_STRIDE + k.
// Stride 68 = 64 + 4 pad floats: keeps every feature row 16B-aligned for
// b128 async copies, and makes bank = (4f + t0) mod 64 so the data-dependent
// knot index t0 no longer pins all 32 lanes to one bank.
#define LDS_STRIDE  68
#define LDS_BYTES   (NFEAT * LDS_STRIDE * 4)   // 139,264 B (< 320 KB/WGP)

// Native clang vector type: accepted by __builtin_nontemporal_load
// (HIP's float4 is a class and is rejected).
typedef float v4f __attribute__((ext_vector_type(4)));

__global__ __launch_bounds__(BLOCK_THREADS)
void spline_layer_kernel(const float* __restrict__ x,
                         const float* __restrict__ coeffs,
                         float* __restrict__ out,
                         int nrows) {
    extern __shared__ float lds_coeffs[];

    const int tid  = threadIdx.x;
    const int lane = tid & 31;
    const int wid  = tid >> 5;

    // ---------------------------------------------------------------
    // Stage coeffs (512 x 64 f32 = 128 KB) into padded LDS using the
    // CDNA5 async global->LDS DMA path (tracked by ASYNCcnt).
    // Each chunk: 4 floats (16 B). 512*16 = 8192 chunks / 256 thr = 32 each.
    // ---------------------------------------------------------------
    const unsigned lds_base = (unsigned)(uintptr_t)(&lds_coeffs[0]);
    for (int c = tid; c < NFEAT * (NKNOT / 4); c += BLOCK_THREADS) {
        const int f  = c >> 4;              // feature row
        const int k4 = (c & 15) << 2;       // starting knot within row
        const float* gsrc = coeffs + f * NKNOT + k4;
        const unsigned ldst = lds_base + ((unsigned)(f * LDS_STRIDE + k4) << 2);
        asm volatile("global_load_async_to_lds_b128 %0, %1, off"
                     :
                     : "v"(ldst), "v"(gsrc)
                     : "memory");
    }
    asm volatile("s_wait_asynccnt 0" ::: "memory");
    __syncthreads();

    // ---------------------------------------------------------------
    // One wave per row: 32 lanes x 4 float4 = 512 features of x.
    // x is single-use -> non-temporal loads (keep L2 for coeffs reuse).
    // ---------------------------------------------------------------
    const int wave_global = blockIdx.x * WAVES_PER_BLOCK + wid;

    #pragma unroll 1
    for (int r = 0; r < ROWS_PER_WAVE; ++r) {
        const int row = wave_global * ROWS_PER_WAVE + r;
        if (row >= nrows) break;

        const v4f* __restrict__ xr =
            reinterpret_cast<const v4f*>(x + (size_t)row * NFEAT);

        float acc = 0.0f;
        #pragma unroll
        for (int i = 0; i < 4; ++i) {
            const int m = i * 32 + lane;                 // float4 index in row
            const v4f v = __builtin_nontemporal_load(&xr[m]);
            const int fbase = m << 2;                    // feature of v[0]
            #pragma unroll
            for (int j = 0; j < 4; ++j) {
                const float t = (v[j] - XMIN_F) * T_SCALE;
                int t0 = (int)floorf(t);
                t0 = t0 < 0 ? 0 : (t0 > (NKNOT - 2) ? (NKNOT - 2) : t0);
                const float w1 = t - (float)t0;
                const float w0 = 1.0f - w1;
                const float* cp = &lds_coeffs[(fbase + j) * LDS_STRIDE + t0];
                const float c0 = cp[0];
                const float c1 = cp[1];
                acc += w0 * c0 + w1 * c1;
            }
        }

        // Wave32 reduction over the 512 features handled by this wave.
        #pragma unroll
        for (int off = 16; off > 0; off >>= 1)
            acc += __shfl_down(acc, off, 32);

        if (lane == 0) out[row] = acc;
    }
}

extern "C" void kernel_launch(void* const* d_in, const int* in_sizes, int n_in,
                              void* d_out, int out_size, void* d_ws, size_t ws_size,
                              hipStream_t stream) {
    const float* x      = (const float*)d_in[0];   // (B, 512) f32
    const float* coeffs = (const float*)d_in[1];   // (512, 64) f32
    float* out          = (float*)d_out;           // (B,) f32

    const int nrows = in_sizes[0] / NFEAT;         // 65536
    const int grid  = (nrows + ROWS_PER_BLOCK - 1) / ROWS_PER_BLOCK;  // 512

    spline_layer_kernel<<<grid, BLOCK_THREADS, LDS_BYTES, stream>>>(
        x, coeffs, out, nrows);
}